// SpatialPropagationBlock_65240553226706
// MI455X (gfx1250) — compile-verified
//
#include <hip/hip_runtime.h>
#include <hip/hip_bf16.h>

// SPN GateRecurrent2dnoind, 4 directions fused via global f32 max atomics.
// Memory-bound (~1.9 GB traffic -> ~80us floor @ 23.3 TB/s). No matmul
// structure -> WMMA inapplicable; CDNA5 paths used: global_load_async_to_lds
// (ASYNCcnt double-buffered staging) + global_atomic_max_num_f32 (L2-resident
// output: 33.5 MB < 192 MB L2).

namespace {

constexpr int kW = 256;           // plane width == height
constexpr int kHW = 256 * 256;    // elements per plane
constexpr int kChunk = 16;        // scan steps staged per chunk
constexpr int kNChunks = 16;      // 256 / 16

__device__ __forceinline__ void async_ld_b128(unsigned lds, const float* g) {
  asm volatile("global_load_async_to_lds_b128 %0, %1, off"
               :: "v"(lds), "v"(g) : "memory");
}
__device__ __forceinline__ void async_ld_b64(unsigned lds, const float* g) {
  asm volatile("global_load_async_to_lds_b64 %0, %1, off"
               :: "v"(lds), "v"(g) : "memory");
}
__device__ __forceinline__ void global_max_f32(float* p, float v) {
  asm volatile("global_atomic_max_num_f32 %0, %1, off"
               :: "v"(p), "v"(v) : "memory");
}
#define WAIT_ASYNC_LE(n) asm volatile("s_wait_asynccnt " #n ::: "memory")

// Gate normalization, matching the reference's single-rounded g/s divides.
// Depends only on staged tile data -> off the serial h-recurrence path.
__device__ __forceinline__ void normalize_gates(float& a, float& g, float& c) {
  const float ssum = fabsf(a) + fabsf(g) + fabsf(c) + 1e-7f;
  if (ssum >= 1.0f) { a = a / ssum; g = g / ssum; c = c / ssum; }
}

// ---------------------------------------------------------------------------
// Vertical directions (dirs 2,3): scan over rows H, cross dim = W (contiguous).
// LDS: tiles[2][4][16*256] floats + hbuf[2][258]  -> 133,136 B dynamic.
// ---------------------------------------------------------------------------
__global__ __launch_bounds__(256) void spn_vertical(
    const float* __restrict__ x, const float* __restrict__ mask,
    float* __restrict__ out) {
  extern __shared__ float4 smem4[];
  float* smem = reinterpret_cast<float*>(smem4);
  const int tid = threadIdx.x;
  const int task = blockIdx.x;          // 0..255
  const int rv = task >> 7;             // 0: fwd, 1: rev
  const int plane = task & 127;         // b*32 + c
  const int b = plane >> 5;
  const int c = plane & 31;
  const int d = 2 + rv;                 // direction index in reference
  const bool rev = (rv != 0);

  const float* gb[4];
  gb[0] = mask + (size_t)plane * kHW;
  gb[1] = x + ((size_t)b * 384 + (size_t)(3 * d + 0) * 32 + c) * kHW;
  gb[2] = x + ((size_t)b * 384 + (size_t)(3 * d + 1) * 32 + c) * kHW;
  gb[3] = x + ((size_t)b * 384 + (size_t)(3 * d + 2) * 32 + c) * kHW;
  float* ob = out + (size_t)plane * kHW;

  const unsigned smem_base = (unsigned)(size_t)smem;  // low 32 bits = LDS offset
  float* hbuf = smem + 2 * 4 * 4096;                  // [2][258], guard zeros

  hbuf[tid + 1] = 0.0f;
  hbuf[258 + tid + 1] = 0.0f;
  if (tid < 2) { hbuf[tid * 257] = 0.0f; hbuf[258 + tid * 257] = 0.0f; }

  auto stage = [&](int ci, int bi) {     // 16 async b128 per wave
    const int h0 = rev ? (240 - 16 * ci) : (16 * ci);
#pragma unroll
    for (int arr = 0; arr < 4; ++arr) {
      const unsigned tb = smem_base + (unsigned)((bi * 4 + arr) * 4096) * 4u;
      const float* bp = gb[arr];
#pragma unroll
      for (int it = 0; it < 4; ++it) {
        const int fv = it * 256 + tid;   // vec4 index in 16x256 tile
        const int r = fv >> 6;           // row within block (memory order)
        const int q = fv & 63;           // vec4 column
        const int s = rev ? (15 - r) : r;  // scan-local slot
        async_ld_b128(tb + (unsigned)(s * 256 + q * 4) * 4u,
                      bp + (size_t)(h0 + r) * kW + q * 4);
      }
    }
  };

  stage(0, 0);
  for (int ci = 0; ci < kNChunks; ++ci) {
    const int bi = ci & 1;
    if (ci + 1 < kNChunks) { stage(ci + 1, bi ^ 1); WAIT_ASYNC_LE(16); }
    else                   { WAIT_ASYNC_LE(0); }
    __syncthreads();
    const float* Tm = smem + (bi * 4 + 0) * 4096;
    const float* T1 = smem + (bi * 4 + 1) * 4096;
    const float* T2 = smem + (bi * 4 + 2) * 4096;
    const float* T3 = smem + (bi * 4 + 3) * 4096;
#pragma unroll 4
    for (int s = 0; s < kChunk; ++s) {
      const int t = ci * kChunk + s;
      const int row = rev ? (255 - t) : t;
      const int idx = s * 256 + tid;
      const float xv = Tm[idx];
      float a = T1[idx], g = T2[idx], cc = T3[idx];
      normalize_gates(a, g, cc);
      float* hc = hbuf + ((t & 1) ? 258 : 0);
      float* hn = hbuf + ((t & 1) ? 0 : 258);
      const float hnew = (1.0f - a - g - cc) * xv +
                         a * hc[tid] + g * hc[tid + 1] + cc * hc[tid + 2];
      hn[tid + 1] = hnew;
      global_max_f32(ob + (size_t)row * kW + tid, hnew);
      __syncthreads();
    }
  }
}

// ---------------------------------------------------------------------------
// Horizontal directions (dirs 0,1): scan over cols W, cross dim = H (strided).
// Stage 16-column blocks transposed into stride-18 padded LDS rows (bank-
// conflict-free column reads, 8B-aligned b64 LDS dests).
// LDS: tiles[2][4][256*18] floats + hbuf[2][258]  -> 149,520 B dynamic.
// ---------------------------------------------------------------------------
__global__ __launch_bounds__(256) void spn_horizontal(
    const float* __restrict__ x, const float* __restrict__ mask,
    float* __restrict__ out) {
  extern __shared__ float4 smem4[];
  float* smem = reinterpret_cast<float*>(smem4);
  const int tid = threadIdx.x;
  const int task = blockIdx.x;
  const int rv = task >> 7;
  const int plane = task & 127;
  const int b = plane >> 5;
  const int c = plane & 31;
  const int d = rv;                     // dirs 0 (fwd), 1 (rev)
  const bool rev = (rv != 0);

  const float* gb[4];
  gb[0] = mask + (size_t)plane * kHW;
  gb[1] = x + ((size_t)b * 384 + (size_t)(3 * d + 0) * 32 + c) * kHW;
  gb[2] = x + ((size_t)b * 384 + (size_t)(3 * d + 1) * 32 + c) * kHW;
  gb[3] = x + ((size_t)b * 384 + (size_t)(3 * d + 2) * 32 + c) * kHW;
  float* ob = out + (size_t)plane * kHW;

  constexpr int kTile = 256 * 18;       // padded tile, floats
  const unsigned smem_base = (unsigned)(size_t)smem;
  float* hbuf = smem + 2 * 4 * kTile;

  hbuf[tid + 1] = 0.0f;
  hbuf[258 + tid + 1] = 0.0f;
  if (tid < 2) { hbuf[tid * 257] = 0.0f; hbuf[258 + tid * 257] = 0.0f; }

  auto stage = [&](int ci, int bi) {     // 32 async b64 per wave
    const int wlo = rev ? (240 - 16 * ci) : (16 * ci);
#pragma unroll
    for (int arr = 0; arr < 4; ++arr) {
      const unsigned tb = smem_base + (unsigned)((bi * 4 + arr) * kTile) * 4u;
      const float* bp = gb[arr];
#pragma unroll
      for (int it = 0; it < 8; ++it) {
        const int fv = it * 256 + tid;   // float-pair index in 256x16 tile
        const int h = fv >> 3;           // source row 0..255
        const int p = fv & 7;            // pair (2 cols) within 16-col block
        async_ld_b64(tb + (unsigned)(h * 18 + p * 2) * 4u,
                     bp + (size_t)h * kW + wlo + p * 2);
      }
    }
  };

  stage(0, 0);
  for (int ci = 0; ci < kNChunks; ++ci) {
    const int bi = ci & 1;
    if (ci + 1 < kNChunks) { stage(ci + 1, bi ^ 1); WAIT_ASYNC_LE(32); }
    else                   { WAIT_ASYNC_LE(0); }
    __syncthreads();
    const int wlo = rev ? (240 - 16 * ci) : (16 * ci);
    const float* Tm = smem + (bi * 4 + 0) * kTile;
    const float* T1 = smem + (bi * 4 + 1) * kTile;
    const float* T2 = smem + (bi * 4 + 2) * kTile;
    const float* T3 = smem + (bi * 4 + 3) * kTile;
#pragma unroll 4
    for (int s = 0; s < kChunk; ++s) {
      const int t = ci * kChunk + s;
      const int js = rev ? (15 - s) : s;  // j == w - wlo for both directions
      const int wcol = wlo + js;          // == rev ? 255 - t : t
      const int idx = tid * 18 + js;
      const float xv = Tm[idx];
      float a = T1[idx], g = T2[idx], cc = T3[idx];
      normalize_gates(a, g, cc);
      float* hc = hbuf + ((t & 1) ? 258 : 0);
      float* hn = hbuf + ((t & 1) ? 0 : 258);
      const float hnew = (1.0f - a - g - cc) * xv +
                         a * hc[tid] + g * hc[tid + 1] + cc * hc[tid + 2];
      hn[tid + 1] = hnew;
      global_max_f32(ob + (size_t)tid * kW + wcol, hnew);
      __syncthreads();
    }
  }
}

// Initialize output to -inf every call (atomic-max accumulator; harness
// poisons d_out and does not restore between replays).
__global__ __launch_bounds__(256) void spn_init_out(float4* __restrict__ out4,
                                                    int n4) {
  const int i = blockIdx.x * blockDim.x + threadIdx.x;
  const float ninf = -__builtin_huge_valf();
  if (i < n4) out4[i] = make_float4(ninf, ninf, ninf, ninf);
}

}  // namespace

extern "C" void kernel_launch(void* const* d_in, const int* in_sizes, int n_in,
                              void* d_out, int out_size, void* d_ws,
                              size_t ws_size, hipStream_t stream) {
  (void)in_sizes; (void)n_in; (void)d_ws; (void)ws_size;
  const float* x = (const float*)d_in[0];     // [4, 384, 256, 256] f32
  const float* mask = (const float*)d_in[1];  // [4,  32, 256, 256] f32
  float* out = (float*)d_out;                 // [4,  32, 256, 256] f32

  const size_t vert_smem = (size_t)(2 * 4 * 4096 + 2 * 258) * sizeof(float);
  const size_t horz_smem = (size_t)(2 * 4 * (256 * 18) + 2 * 258) * sizeof(float);
  hipFuncSetAttribute(reinterpret_cast<const void*>(spn_vertical),
                      hipFuncAttributeMaxDynamicSharedMemorySize,
                      (int)vert_smem);
  hipFuncSetAttribute(reinterpret_cast<const void*>(spn_horizontal),
                      hipFuncAttributeMaxDynamicSharedMemorySize,
                      (int)horz_smem);

  const int n4 = out_size / 4;
  spn_init_out<<<(n4 + 255) / 256, 256, 0, stream>>>((float4*)out, n4);
  spn_horizontal<<<256, 256, horz_smem, stream>>>(x, mask, out);
  spn_vertical<<<256, 256, vert_smem, stream>>>(x, mask, out);
}